// DKEPooling_59176059405035
// MI455X (gfx1250) — compile-verified
//
#include <hip/hip_runtime.h>
#include <hip/hip_bf16.h>

typedef __attribute__((ext_vector_type(2))) float v2f;
typedef __attribute__((ext_vector_type(8))) float v8f;

// ---------------------------------------------------------------------------
// Stage 1: per-graph scalar stats -> SNR scale + noise scalar mean
// ---------------------------------------------------------------------------
__global__ __launch_bounds__(256)
void stats_kernel(const float* __restrict__ feat, const float* __restrict__ noise,
                  float* __restrict__ scl, float* __restrict__ nmn,
                  int nd, float snr_pow) {
  int b = blockIdx.x;
  int t = threadIdx.x;
  const float* f  = feat  + (long long)b * nd;
  const float* nz = noise + (long long)b * nd;
  double sf = 0.0, sf2 = 0.0, sn = 0.0, sn2 = 0.0;
  for (int i = t; i < nd; i += 256) {
    double x = (double)f[i];
    double y = (double)nz[i];
    sf += x; sf2 += x * x; sn += y; sn2 += y * y;
  }
  __shared__ double sh[4][256];
  sh[0][t] = sf; sh[1][t] = sf2; sh[2][t] = sn; sh[3][t] = sn2;
  __syncthreads();
  for (int s = 128; s > 0; s >>= 1) {
    if (t < s) {
      sh[0][t] += sh[0][t + s]; sh[1][t] += sh[1][t + s];
      sh[2][t] += sh[2][t + s]; sh[3][t] += sh[3][t + s];
    }
    __syncthreads();
  }
  if (t == 0) {
    double inv = 1.0 / (double)nd;
    double gmf = sh[0][0] * inv;
    double gmn = sh[2][0] * inv;
    double tvar = sh[1][0] * inv - gmf * gmf;
    double nvar = sh[3][0] * inv - gmn * gmn;
    scl[b] = (float)sqrt((tvar / (double)snr_pow) / nvar);
    nmn[b] = (float)gmn;
  }
}

// ---------------------------------------------------------------------------
// Stage 2: feat' = feat + (noise - nmean_b) * scale_b ; column means; center
// ---------------------------------------------------------------------------
__global__ __launch_bounds__(256)
void perturb_kernel(const float* __restrict__ feat, const float* __restrict__ noise,
                    const float* __restrict__ scl, const float* __restrict__ nmn,
                    float* __restrict__ Fp, int nd) {
  int b = blockIdx.y;
  long long idx = (long long)b * nd + (long long)blockIdx.x * 256 + threadIdx.x;
  Fp[idx] = feat[idx] + (noise[idx] - nmn[b]) * scl[b];
}

__global__ __launch_bounds__(256)
void mean_kernel(const float* __restrict__ Fp, float* __restrict__ mean, int n, int d) {
  int b = blockIdx.x;
  int j = threadIdx.x;
  const float* base = Fp + (long long)b * n * d;
  float s = 0.0f;
  for (int i = 0; i < n; ++i) s += base[(long long)i * d + j];
  mean[(long long)b * d + j] = s / (float)n;
}

__global__ __launch_bounds__(256)
void center_kernel(float* __restrict__ Fp, const float* __restrict__ mean, int n, int d) {
  int row = blockIdx.x;          // b*n + i
  int j   = threadIdx.x;
  int b   = row / n;
  Fp[(long long)row * d + j] -= mean[(long long)b * d + j];
}

// ---------------------------------------------------------------------------
// Batched fp32 WMMA GEMM:  D[b] = f(A[b] @ B[b])
//   mode 0: D = alpha * (A@B)
//   mode 1: D = 1.5*I - 0.5*(A@B)      (Newton-Schulz ZY step, fused)
// 128x128 block tile, 256 threads = 8 waves, each wave a 32x64 slab
// (2 A-frags x 4 B-frags -> 8 v_wmma_f32_16x16x4_f32 per K-step).
// B tile is stored TRANSPOSED in LDS so every fragment is a contiguous,
// 8B-aligned float2 -> single ds_load_b64 into an even VGPR pair.
// Row pad = 18 dwords keeps b64 alignment and conflict-free column reads.
// ---------------------------------------------------------------------------
#define BM 128
#define BN 128
#define BK 16
#define LDP 18

__global__ __launch_bounds__(256)
void wmma_gemm_batched(const float* __restrict__ A, long long sAb, int sAm, int sAk,
                       const float* __restrict__ Bm, long long sBb, int sBk, int sBn,
                       float* __restrict__ Dm, long long sDb,
                       int N, int K, float alpha, int mode) {
  __shared__ __align__(16) float As[BM][LDP];
  __shared__ __align__(16) float Bst[BN][LDP];   // B^T tile: [n][k]

  int b  = blockIdx.z;
  int m0 = blockIdx.y * BM;
  int n0 = blockIdx.x * BN;
  const float* Ab = A  + (long long)b * sAb;
  const float* Bb = Bm + (long long)b * sBb;
  float*       Db = Dm + (long long)b * sDb;

  int tid  = threadIdx.x;
  int wid  = tid >> 5;
  int lane = tid & 31;
  int mr   = lane & 15;          // M/N index within 16
  int h    = lane >> 4;          // half-wave: K pair select / M+8 in C layout
  int wm   = (wid >> 1) * 32;    // 4 wave-rows of 32
  int wn   = (wid & 1) * 64;     // 2 wave-cols of 64

  v8f acc[2][4];
#pragma unroll
  for (int im = 0; im < 2; ++im)
#pragma unroll
    for (int in = 0; in < 4; ++in) acc[im][in] = (v8f){};

  for (int k0 = 0; k0 < K; k0 += BK) {
    // stage A (BMxBK = 2048 floats): coalesced along k when sAk==1
#pragma unroll
    for (int i = 0; i < 8; ++i) {
      int e = tid + i * 256;
      int m = e >> 4, kk = e & 15;
      As[m][kk] = Ab[(long long)(m0 + m) * sAm + (long long)(k0 + kk) * sAk];
    }
    // stage B transposed (BKxBN read, [n][k] written): coalesced along n
#pragma unroll
    for (int i = 0; i < 8; ++i) {
      int e = tid + i * 256;
      int nn = e & 127, kk = e >> 7;
      Bst[nn][kk] = Bb[(long long)(k0 + kk) * sBk + (long long)(n0 + nn) * sBn];
    }
    __syncthreads();
#pragma unroll
    for (int kk = 0; kk < BK; kk += 4) {
      int kc = kk + 2 * h;       // even -> 8B aligned with LDP=18
      v2f a0 = *(const v2f*)&As[wm + mr][kc];
      v2f a1 = *(const v2f*)&As[wm + 16 + mr][kc];
      v2f b0 = *(const v2f*)&Bst[wn + mr][kc];
      v2f b1 = *(const v2f*)&Bst[wn + 16 + mr][kc];
      v2f b2 = *(const v2f*)&Bst[wn + 32 + mr][kc];
      v2f b3 = *(const v2f*)&Bst[wn + 48 + mr][kc];
      acc[0][0] = __builtin_amdgcn_wmma_f32_16x16x4_f32(false, a0, false, b0, (short)0, acc[0][0], false, false);
      acc[0][1] = __builtin_amdgcn_wmma_f32_16x16x4_f32(false, a0, false, b1, (short)0, acc[0][1], false, false);
      acc[0][2] = __builtin_amdgcn_wmma_f32_16x16x4_f32(false, a0, false, b2, (short)0, acc[0][2], false, false);
      acc[0][3] = __builtin_amdgcn_wmma_f32_16x16x4_f32(false, a0, false, b3, (short)0, acc[0][3], false, false);
      acc[1][0] = __builtin_amdgcn_wmma_f32_16x16x4_f32(false, a1, false, b0, (short)0, acc[1][0], false, false);
      acc[1][1] = __builtin_amdgcn_wmma_f32_16x16x4_f32(false, a1, false, b1, (short)0, acc[1][1], false, false);
      acc[1][2] = __builtin_amdgcn_wmma_f32_16x16x4_f32(false, a1, false, b2, (short)0, acc[1][2], false, false);
      acc[1][3] = __builtin_amdgcn_wmma_f32_16x16x4_f32(false, a1, false, b3, (short)0, acc[1][3], false, false);
    }
    __syncthreads();
  }

#pragma unroll
  for (int im = 0; im < 2; ++im) {
#pragma unroll
    for (int r = 0; r < 8; ++r) {
      int row = m0 + wm + 16 * im + 8 * h + r;   // C layout: VGPR r -> M=r / M=8+r
      float* drow = Db + (long long)row * N;
#pragma unroll
      for (int in = 0; in < 4; ++in) {
        int col = n0 + wn + 16 * in + mr;
        float v = acc[im][in][r];
        if (mode == 1) {
          v = -0.5f * v + ((row == col) ? 1.5f : 0.0f);
        } else {
          v *= alpha;
        }
        drow[col] = v;
      }
    }
  }
}

// ---------------------------------------------------------------------------
// trace, A/ZY0 init, final GEMV
// ---------------------------------------------------------------------------
__global__ __launch_bounds__(256)
void trace_kernel(const float* __restrict__ cov, float* __restrict__ tr, int d) {
  int b = blockIdx.x;
  int t = threadIdx.x;
  __shared__ float sh[256];
  sh[t] = cov[(long long)b * d * d + (long long)t * d + t];
  __syncthreads();
  for (int s = 128; s > 0; s >>= 1) {
    if (t < s) sh[t] += sh[t + s];
    __syncthreads();
  }
  if (t == 0) tr[b] = sh[0];
}

__global__ __launch_bounds__(256)
void init_A_ZY(const float* __restrict__ cov, const float* __restrict__ tr,
               float* __restrict__ Amat, float* __restrict__ ZY, int d) {
  int bi = blockIdx.x;            // b*d + i
  int b  = bi / d;
  int i  = bi % d;
  int j  = threadIdx.x;
  long long idx = (long long)b * d * d + (long long)i * d + j;
  float a = cov[idx] / tr[b];
  Amat[idx] = a;
  ZY[idx]   = -0.5f * a + ((i == j) ? 1.5f : 0.0f);   // 0.5*(3I - A)
}

__global__ __launch_bounds__(256)
void out_kernel(const float* __restrict__ YZY, const float* __restrict__ mean,
                const float* __restrict__ tr, float* __restrict__ out, int d) {
  int wid  = threadIdx.x >> 5;
  int lane = threadIdx.x & 31;
  int row  = blockIdx.x * 8 + wid;            // row in [0, B*d)
  int b    = row / d;
  int i    = row % d;
  const float* yrow = YZY + (long long)b * d * d + (long long)i * d;
  const float* mrow = mean + (long long)b * d;
  float s = 0.0f;
  for (int j = lane; j < d; j += 32) s += yrow[j] * mrow[j];
  for (int off = 16; off > 0; off >>= 1) s += __shfl_down(s, off, 32);
  if (lane == 0) out[row] = s * sqrtf(tr[b]);
}

// ---------------------------------------------------------------------------
extern "C" void kernel_launch(void* const* d_in, const int* in_sizes, int n_in,
                              void* d_out, int out_size, void* d_ws, size_t ws_size,
                              hipStream_t stream) {
  (void)n_in; (void)ws_size; (void)out_size;
  const float* feat  = (const float*)d_in[1];
  const float* noise = (const float*)d_in[2];

  const int B  = in_sizes[0];            // 128
  const int d  = 256;                    // feature dim (reference setup)
  const long long total = (long long)in_sizes[1];
  const int n  = (int)(total / ((long long)B * d));   // 128 nodes/graph
  const int nd = n * d;
  const long long dd = (long long)d * d;

  float* W = (float*)d_ws;
  long long off = 0;
  float* Fp   = W + off; off += (long long)B * nd;   // perturbed / centered feats
  float* mean = W + off; off += (long long)B * d;
  float* scl  = W + off; off += B;
  float* nmn  = W + off; off += B;
  float* tr   = W + off; off += B;
  off = (off + 255) & ~255LL;
  float* M0 = W + off; off += (long long)B * dd;
  float* M1 = W + off; off += (long long)B * dd;
  float* M2 = W + off; off += (long long)B * dd;
  float* M3 = W + off; off += (long long)B * dd;

  // 1) per-graph stats -> SNR scale (10^(15/10))
  stats_kernel<<<B, 256, 0, stream>>>(feat, noise, scl, nmn, nd, 31.6227766016838f);
  // 2) perturb + per-graph column mean + center
  perturb_kernel<<<dim3(nd / 256, B), 256, 0, stream>>>(feat, noise, scl, nmn, Fp, nd);
  mean_kernel<<<B, d, 0, stream>>>(Fp, mean, n, d);
  center_kernel<<<B * n, d, 0, stream>>>(Fp, mean, n, d);

  dim3 gg(d / BN, d / BM, B);
  // 3) cov = diff^T @ diff / (n-1)  (A strided as diff^T)
  wmma_gemm_batched<<<gg, 256, 0, stream>>>(Fp, (long long)nd, 1, d,
                                            Fp, (long long)nd, d, 1,
                                            M0, dd, d, n,
                                            1.0f / (float)(n - 1), 0);
  trace_kernel<<<B, d, 0, stream>>>(M0, tr, d);
  init_A_ZY<<<B * d, d, 0, stream>>>(M0, tr, M1, M2, d);   // A -> M1, ZY0 -> M2

  // 4) Y = A @ ZY0 -> M0 ; Z = ZY0 (M2)
  wmma_gemm_batched<<<gg, 256, 0, stream>>>(M1, dd, d, 1, M2, dd, d, 1,
                                            M0, dd, d, d, 1.0f, 0);
  float* Y = M0; float* Z = M2; float* S1 = M1; float* S2 = M3;

  // 5) Newton-Schulz iterations (ITERN=5 -> 3 loop iterations)
  for (int it = 0; it < 3; ++it) {
    wmma_gemm_batched<<<gg, 256, 0, stream>>>(Z, dd, d, 1, Y, dd, d, 1,
                                              S1, dd, d, d, 1.0f, 1);  // ZY = 1.5I-0.5*Z@Y
    wmma_gemm_batched<<<gg, 256, 0, stream>>>(Y, dd, d, 1, S1, dd, d, 1,
                                              S2, dd, d, d, 1.0f, 0);  // Ynew
    wmma_gemm_batched<<<gg, 256, 0, stream>>>(S1, dd, d, 1, Z, dd, d, 1,
                                              Y, dd, d, d, 1.0f, 0);   // Znew (old Y buf)
    float* newY = S2; float* newZ = Y; float* freed = Z;
    Y = newY; Z = newZ; S2 = freed;    // 4 distinct buffers maintained
  }

  // 6) YZY = Y @ (1.5I - 0.5*Z@Y)
  wmma_gemm_batched<<<gg, 256, 0, stream>>>(Z, dd, d, 1, Y, dd, d, 1,
                                            S1, dd, d, d, 1.0f, 1);
  wmma_gemm_batched<<<gg, 256, 0, stream>>>(Y, dd, d, 1, S1, dd, d, 1,
                                            S2, dd, d, d, 1.0f, 0);

  // 7) out = sqrt(tr) * (YZY @ mean)
  out_kernel<<<(B * d) / 8, 256, 0, stream>>>(S2, mean, tr, (float*)d_out, d);
}